// LigerLMHeadCE_22101901705771
// MI455X (gfx1250) — compile-verified
//
#include <hip/hip_runtime.h>
#include <hip/hip_bf16.h>
#include <math.h>

// Problem constants (match reference)
#define H   2048
#define BT  2048
#define V   128000

// Tiling
#define TB   16      // tokens per workgroup
#define VB   512     // logit columns per workgroup
#define NVB  (V / VB)        // 250 V-blocks
#define NTB  (BT / TB)       // 128 token blocks
#define KC   32      // K chunk staged in LDS
#define KP   36      // padded row stride (dwords): conflict-free + 16B aligned
#define NCHUNK (H / KC)      // 64
#define LSP  513     // padded logits row stride

typedef float v2f __attribute__((ext_vector_type(2)));
typedef float v8f __attribute__((ext_vector_type(8)));

// ---------------------------------------------------------------------------
// Async W staging: each thread issues 16 global_load_async_to_lds_b128, i.e.
// 16 ASYNCcnt increments per wave per chunk. No VGPR round trip, no loadcnt
// stalls; completion enforced with s_wait_asynccnt before the barrier.
// ---------------------------------------------------------------------------
__device__ __forceinline__ void issue_w_async(const float* wbase /*uniform: wrow+kc*/,
                                              const float* wbuf  /*LDS dst buffer*/,
                                              int tid)
{
    const int rbase = tid >> 3;            // 0..31 (+32 per step)
    const int c4    = (tid & 7) << 2;      // 0,4,...,28
    unsigned lds = (unsigned)(unsigned long long)(wbuf + rbase * KP + c4);
    unsigned gof = (unsigned)(((unsigned)rbase * H + (unsigned)c4) * sizeof(float));
    #pragma unroll
    for (int i = 0; i < 16; ++i) {
        asm volatile("global_load_async_to_lds_b128 %0, %1, %2"
                     :: "v"(lds), "v"(gof), "s"(wbase)
                     : "memory");
        lds += 32u * KP * (unsigned)sizeof(float);   // +32 rows in LDS
        gof += 32u * H  * (unsigned)sizeof(float);   // +32 rows in W
    }
}

__device__ __forceinline__ void stage_x(const float* xbase /*xrow+kc*/,
                                        float* xbuf, int tid)
{
    if (tid < 128) {
        const int r  = tid >> 3;
        const int c4 = (tid & 7) << 2;
        const float4 v = *(const float4*)(xbase + (size_t)r * H + c4);
        *(float4*)(&xbuf[r * KP + c4]) = v;
    }
}

__device__ __forceinline__ void mma_chunk(const float* xsb, const float* wlb,
                                          int arow, int halfsel, int wave,
                                          v8f acc[4])
{
    #pragma unroll
    for (int k4 = 0; k4 < KC; k4 += 4) {
        const v2f a = *(const v2f*)(&xsb[arow * KP + k4 + halfsel]);
        #pragma unroll
        for (int t = 0; t < 4; ++t) {
            const int col = ((wave << 2) + t) * 16 + arow;
            const v2f bf  = *(const v2f*)(&wlb[col * KP + k4 + halfsel]);
            acc[t] = __builtin_amdgcn_wmma_f32_16x16x4_f32(
                false, a, false, bf, (short)0, acc[t], false, false);
        }
    }
}

// ---------------------------------------------------------------------------
// Kernel 1: 16x512 logit tile via v_wmma_f32_16x16x4_f32 with double-buffered
// async-to-LDS W staging, fused online softmax partials.
// grid = (NTB, NVB): token-block fast axis -> all 128 token blocks of one 4MB
// W slice are co-resident, so W streams from HBM ~once (L2 captures reuse).
// ---------------------------------------------------------------------------
__global__ __launch_bounds__(256)
void ce_logits_partials(const float* __restrict__ x,
                        const int*   __restrict__ y,
                        const float* __restrict__ W,
                        const float* __restrict__ bvec,
                        float* __restrict__ pmax,   // [BT][NVB]
                        float* __restrict__ psum,   // [BT][NVB]
                        float* __restrict__ tlog)   // [BT]
{
    __shared__ float xs0[TB * KP], xs1[TB * KP];     // x chunk double buffer
    __shared__ float wl0[VB * KP], wl1[VB * KP];     // W chunk double buffer
    __shared__ float pm[16 * 16];
    __shared__ float ps[16 * 16];

    const int tb   = blockIdx.x;       // token block
    const int vb   = blockIdx.y;       // V block
    const int tid  = threadIdx.x;
    const int wave = tid >> 5;
    const int lane = tid & 31;

    const float* xrow = x + (size_t)tb * TB * H;
    const float* wrow = W + (size_t)vb * VB * H;

    v8f acc[4];
    #pragma unroll
    for (int t = 0; t < 4; ++t)
        #pragma unroll
        for (int r = 0; r < 8; ++r) acc[t][r] = 0.0f;

    const int arow    = lane & 15;
    const int halfsel = (lane < 16) ? 0 : 2;   // f32 frag: lanes16-31 hold K+2,K+3

    // Prologue: stage chunk 0
    stage_x(xrow, xs0, tid);
    issue_w_async(wrow, wl0, tid);

    for (int i = 0; i < NCHUNK; ++i) {
        const float* xcur = (i & 1) ? xs1 : xs0;
        const float* wcur = (i & 1) ? wl1 : wl0;
        if (i + 1 < NCHUNK) {
            float* xnxt = (i & 1) ? xs0 : xs1;
            float* wnxt = (i & 1) ? wl0 : wl1;
            stage_x(xrow + (i + 1) * KC, xnxt, tid);
            issue_w_async(wrow + (i + 1) * KC, wnxt, tid);
            // Only the 16 just-issued ops may remain outstanding ->
            // chunk i's (older, in-order) async loads have landed.
            asm volatile("s_wait_asynccnt 0x10" ::: "memory");
        } else {
            asm volatile("s_wait_asynccnt 0x0" ::: "memory");
        }
        __syncthreads();
        mma_chunk(xcur, wcur, arow, halfsel, wave, acc);
        __syncthreads();
    }

    // Add bias, dump 16x512 logits to LDS (alias W buffer 0: 32.8KB <= 73.7KB)
    float* ls = wl0;
    #pragma unroll
    for (int t = 0; t < 4; ++t) {
        const int ncol  = ((wave << 2) + t) * 16 + arow;
        const float bia = bvec[(size_t)vb * VB + ncol];
        const int rbase = (lane < 16) ? 0 : 8;   // C layout: lanes16-31 hold M=8..15
        #pragma unroll
        for (int r = 0; r < 8; ++r)
            ls[(rbase + r) * LSP + ncol] = acc[t][r] + bia;
    }
    __syncthreads();

    // Per-token max / sum-exp over this 512-col block (token = tid&15,
    // column group of 32 = tid>>4), plus target-logit pick.
    {
        const int t   = tid & 15;
        const int g   = tid >> 4;
        const int tok = tb * TB + t;
        const int tgt_local = y[tok] - vb * VB;
        float m = -INFINITY, s = 0.0f;
        const int c0 = g << 5;
        for (int c = c0; c < c0 + 32; ++c) {
            const float v  = ls[t * LSP + c];
            const float nm = fmaxf(m, v);
            s = s * __expf(m - nm) + __expf(v - nm);
            m = nm;
            if (c == tgt_local) tlog[tok] = v;
        }
        pm[t * 16 + g] = m;
        ps[t * 16 + g] = s;
    }
    __syncthreads();

    if (tid < 16) {
        const int t = tid;
        float m = -INFINITY, s = 0.0f;
        #pragma unroll
        for (int g = 0; g < 16; ++g) {
            const float mg = pm[t * 16 + g];
            const float sg = ps[t * 16 + g];
            const float nm = fmaxf(m, mg);
            s = s * __expf(m - nm) + sg * __expf(mg - nm);
            m = nm;
        }
        const int tok = tb * TB + t;
        pmax[(size_t)tok * NVB + vb] = m;
        psum[(size_t)tok * NVB + vb] = s;
    }
}

// ---------------------------------------------------------------------------
// Kernel 2: merge NVB (max,sumexp) partials per token -> nll[token]
// ---------------------------------------------------------------------------
__global__ __launch_bounds__(256)
void ce_merge(const float* __restrict__ pmax,
              const float* __restrict__ psum,
              const float* __restrict__ tlog,
              float* __restrict__ nll)
{
    __shared__ float sm[256];
    __shared__ float ss[256];
    const int tok = blockIdx.x;
    const int tid = threadIdx.x;

    float m = -INFINITY, s = 0.0f;
    if (tid < NVB) {
        m = pmax[(size_t)tok * NVB + tid];
        s = psum[(size_t)tok * NVB + tid];
    }
    sm[tid] = m; ss[tid] = s;
    __syncthreads();

    for (int off = 128; off > 0; off >>= 1) {
        if (tid < off) {
            const float m1 = sm[tid], s1 = ss[tid];
            const float m2 = sm[tid + off], s2 = ss[tid + off];
            const float nm = fmaxf(m1, m2);
            float ns = 0.0f;
            if (nm != -INFINITY)
                ns = s1 * __expf(m1 - nm) + s2 * __expf(m2 - nm);
            sm[tid] = nm; ss[tid] = ns;
        }
        __syncthreads();
    }
    if (tid == 0)
        nll[tok] = -tlog[tok] + sm[0] + __logf(ss[0]);
}

// ---------------------------------------------------------------------------
// Kernel 3: mean over BT tokens -> d_out[0]
// ---------------------------------------------------------------------------
__global__ __launch_bounds__(256)
void ce_final(const float* __restrict__ nll, float* __restrict__ out)
{
    __shared__ float red[256];
    const int tid = threadIdx.x;
    float s = 0.0f;
    for (int i = tid; i < BT; i += 256) s += nll[i];
    red[tid] = s;
    __syncthreads();
    for (int off = 128; off > 0; off >>= 1) {
        if (tid < off) red[tid] += red[tid + off];
        __syncthreads();
    }
    if (tid == 0) out[0] = red[0] * (1.0f / (float)BT);
}

// ---------------------------------------------------------------------------
extern "C" void kernel_launch(void* const* d_in, const int* in_sizes, int n_in,
                              void* d_out, int out_size, void* d_ws, size_t ws_size,
                              hipStream_t stream)
{
    const float* x    = (const float*)d_in[0];
    const int*   y    = (const int*)  d_in[1];
    const float* W    = (const float*)d_in[2];
    const float* bvec = (const float*)d_in[3];
    float* out = (float*)d_out;

    // Workspace carve-up (~4.1 MB)
    float* ws   = (float*)d_ws;
    float* pmax = ws;                               // BT*NVB
    float* psum = pmax + (size_t)BT * NVB;          // BT*NVB
    float* tlog = psum + (size_t)BT * NVB;          // BT
    float* nll  = tlog + BT;                        // BT

    dim3 grid1(NTB, NVB);   // token-block fast axis -> W slice L2 reuse
    ce_logits_partials<<<grid1, 256, 0, stream>>>(x, y, W, bvec, pmax, psum, tlog);
    ce_merge<<<dim3(BT), 256, 0, stream>>>(pmax, psum, tlog, nll);
    ce_final<<<dim3(1), 256, 0, stream>>>(nll, out);
}